// MIPNetwork_28424093565379
// MI455X (gfx1250) — compile-verified
//
#include <hip/hip_runtime.h>
#include <math.h>

// MIPNetwork for MI455X (gfx1250). fp32 throughout (matches reference).
// Dense layers use V_WMMA_F32_16X16X4_F32 (wave32 fp32 matrix op).
// SpMM edges via per-wave atomic scatter (tables are L2-resident).
//
// d_in layout (top-level dict insertion order, params flattened jax-style
// with sorted keys: cu,out,qc,qo,vu each b1,b2,w1,w2):
//   0 const_values(16384) 1 objective_multipliers(65536) 2 integer_mask(65536, unused)
//   3 edge_vals(524288)   4 edge_var(int,524288)         5 edge_const(int,524288)
//   6.. params (20 arrays)
//
// Workspace usage (~311 MB): V, C, Q, H, left, const_msg, var_msg, scalers.

#define FMDIM 128
#define OUTB  4
#define NPASS 3

typedef __attribute__((ext_vector_type(2))) float v2f;
typedef __attribute__((ext_vector_type(8))) float v8f;

enum { ACT_NONE = 0, ACT_RELU = 1, ACT_SIG = 2 };

__device__ __forceinline__ float sigf(float x) { return 1.0f / (1.0f + expf(-x)); }

// ---------------------------------------------------------------- utilities
__global__ __launch_bounds__(256) void k_fill(float* __restrict__ p, int n, float v) {
  int i = blockIdx.x * 256 + threadIdx.x;
  if (i < n) p[i] = v;
}

__global__ __launch_bounds__(256) void k_scalers(const float* __restrict__ ev,
                                                 const int* __restrict__ econ,
                                                 const int* __restrict__ evar,
                                                 float* __restrict__ sc,
                                                 float* __restrict__ sv, int E) {
  int e = blockIdx.x * 256 + threadIdx.x;
  if (e < E) {
    float w = ev[e];
    atomicAdd(&sc[econ[e]], w);
    atomicAdd(&sv[evar[e]], w);
  }
}

// ------------------------------------------------------- WMMA fp32 GEMM N=128
// Y[M x 128] = epi( X[M x K] @ W[K x 128] + bias ).
// One wave -> 16 rows x 128 cols (8 accumulators). Block = 8 waves = 128 rows.
__global__ __launch_bounds__(256) void k_gemm_wmma(
    const float* __restrict__ X, int ldx, int K,
    const float* __restrict__ W, const float* __restrict__ bias,
    float* __restrict__ Y, int ldy, int act,
    const float* __restrict__ rowScale,                 // nullable: y *= rs[m] (after act)
    const float* __restrict__ Res, int ldr, float resCoef) // nullable: y += c*Res[m][n]
{
  const int wave = threadIdx.x >> 5;
  const int lane = threadIdx.x & 31;
  const int half = lane >> 4;
  const int l16  = lane & 15;
  const int m0   = (blockIdx.x * 8 + wave) * 16;

  v8f acc[8];
#pragma unroll
  for (int nt = 0; nt < 8; ++nt) {
    float bv = bias[nt * 16 + l16];   // bias depends on column only
#pragma unroll
    for (int v = 0; v < 8; ++v) acc[nt][v] = bv;
  }

  const float* xrow = X + (size_t)(m0 + l16) * ldx;
  for (int k0 = 0; k0 < K; k0 += 4) {
    const int ka = k0 + 2 * half;            // A: lanes 0-15 hold K={0,1}, 16-31 K={2,3}
    v2f a;
    a.x = xrow[ka];
    a.y = xrow[ka + 1];
    const float* wr = W + (size_t)ka * FMDIM;
#pragma unroll
    for (int nt = 0; nt < 8; ++nt) {
      v2f b;
      b.x = wr[nt * 16 + l16];               // row k=ka
      b.y = wr[FMDIM + nt * 16 + l16];       // row k=ka+1
      acc[nt] = __builtin_amdgcn_wmma_f32_16x16x4_f32(
          false, a, false, b, (short)0, acc[nt], false, false);
    }
  }

#pragma unroll
  for (int nt = 0; nt < 8; ++nt) {
#pragma unroll
    for (int v = 0; v < 8; ++v) {
      const int m = m0 + 8 * half + v;       // C/D: lanes 16-31 hold M=8..15
      const int n = nt * 16 + l16;
      float y = acc[nt][v];
      if (Res)             y += resCoef * Res[(size_t)m * ldr + n];
      if (act == ACT_RELU) y = fmaxf(y, 0.0f);
      else if (act == ACT_SIG) y = sigf(y);
      if (rowScale)        y *= rowScale[m];
      Y[(size_t)m * ldy + n] = y;
    }
  }
}

// ------------------------------------------------ per-row layernorm + relu
__global__ __launch_bounds__(256) void k_norm_relu(float* __restrict__ H, int rows) {
  const int row  = blockIdx.x * 8 + (threadIdx.x >> 5);
  const int lane = threadIdx.x & 31;
  if (row >= rows) return;
  float4 x = ((const float4*)(H + (size_t)row * FMDIM))[lane];
  float s = x.x + x.y + x.z + x.w;
  float q = x.x * x.x + x.y * x.y + x.z * x.z + x.w * x.w;
#pragma unroll
  for (int m = 16; m >= 1; m >>= 1) {
    s += __shfl_xor(s, m, 32);
    q += __shfl_xor(q, m, 32);
  }
  const float mean = s * (1.0f / 128.0f);
  const float var  = q * (1.0f / 128.0f) - mean * mean;
  const float r    = rsqrtf(var + 1e-5f);
  x.x = fmaxf(0.0f, (x.x - mean) * r);
  x.y = fmaxf(0.0f, (x.y - mean) * r);
  x.z = fmaxf(0.0f, (x.z - mean) * r);
  x.w = fmaxf(0.0f, (x.w - mean) * r);
  ((float4*)(H + (size_t)row * FMDIM))[lane] = x;
}

// ------------------------------------------------ SpMM scatter (wave/edge)
__global__ __launch_bounds__(256) void k_spmm(const float* __restrict__ ev,
                                              const int* __restrict__ src,
                                              const int* __restrict__ dst,
                                              const float* __restrict__ Xs, int lds_,
                                              float* __restrict__ Yd, int ldd, int E) {
  const int e    = blockIdx.x * 8 + (threadIdx.x >> 5);
  const int lane = threadIdx.x & 31;
  if (e >= E) return;
  const float w = ev[e];
  const int s = src[e], d = dst[e];
  const float4 xv = ((const float4*)(Xs + (size_t)s * lds_))[lane];
  float* yp = Yd + (size_t)d * ldd + lane * 4;
  atomicAdd(yp + 0, w * xv.x);
  atomicAdd(yp + 1, w * xv.y);
  atomicAdd(yp + 2, w * xv.z);
  atomicAdd(yp + 3, w * xv.w);
}

// const_grad: var_msg[v][384+f] += ev * (left[c][f] > cv[c] ? 1 : 0)
__global__ __launch_bounds__(256) void k_spmm_act(const float* __restrict__ ev,
                                                  const int* __restrict__ econ,
                                                  const int* __restrict__ evar,
                                                  const float* __restrict__ left,
                                                  const float* __restrict__ cv,
                                                  float* __restrict__ vmsg, int E) {
  const int e    = blockIdx.x * 8 + (threadIdx.x >> 5);
  const int lane = threadIdx.x & 31;
  if (e >= E) return;
  const float w = ev[e];
  const int c = econ[e], v = evar[e];
  const float t = cv[c];
  const float4 l = ((const float4*)(left + (size_t)c * FMDIM))[lane];
  float* yp = vmsg + (size_t)v * 640 + 384 + lane * 4;
  atomicAdd(yp + 0, (l.x > t) ? w : 0.0f);
  atomicAdd(yp + 1, (l.y > t) ? w : 0.0f);
  atomicAdd(yp + 2, (l.z > t) ? w : 0.0f);
  atomicAdd(yp + 3, (l.w > t) ? w : 0.0f);
}

// const_msg = [C | relu(left-cv)/sc | relu(cv-left)/sc]
__global__ __launch_bounds__(256) void k_cmsg(float* __restrict__ cmsg,
                                              const float* __restrict__ C,
                                              const float* __restrict__ left,
                                              const float* __restrict__ cv,
                                              const float* __restrict__ sc, int nCon) {
  const int i = blockIdx.x * 256 + threadIdx.x;
  if (i >= nCon * FMDIM) return;
  const int c = i >> 7, f = i & 127;
  const float l = left[(size_t)c * FMDIM + f];
  const float t = cv[c];
  const float s = sc[c];
  float* row = cmsg + (size_t)c * 384;
  row[f]         = C[(size_t)c * FMDIM + f];
  row[128 + f]   = fmaxf(l - t, 0.0f) / s;
  row[256 + f]   = fmaxf(t - l, 0.0f) / s;
}

// var_msg init: [V | 0 | (qo writes here) | 0 | om-broadcast]
__global__ __launch_bounds__(256) void k_vmsg_init(float* __restrict__ vmsg,
                                                   const float* __restrict__ V,
                                                   const float* __restrict__ om, int nVar) {
  const int i = blockIdx.x * 256 + threadIdx.x;
  if (i >= nVar * FMDIM) return;
  const int v = i >> 7, f = i & 127;
  float* row = vmsg + (size_t)v * 640;
  row[f]         = V[(size_t)v * FMDIM + f];
  row[128 + f]   = 0.0f;
  row[384 + f]   = 0.0f;
  row[512 + f]   = om[v];
}

// c2v /= scalers_v
__global__ __launch_bounds__(256) void k_c2v_div(float* __restrict__ vmsg,
                                                 const float* __restrict__ sv, int nVar) {
  const int i = blockIdx.x * 256 + threadIdx.x;
  if (i >= nVar * FMDIM) return;
  const int v = i >> 7, f = i & 127;
  vmsg[(size_t)v * 640 + 128 + f] /= sv[v];
}

// final 128 -> 4 head: sigmoid to outSig, raw (optionally) to outRaw
__global__ __launch_bounds__(256) void k_out_head(const float* __restrict__ H,
                                                  const float* __restrict__ w2,
                                                  const float* __restrict__ b2,
                                                  float* __restrict__ outSig,
                                                  float* __restrict__ outRaw, int nVar) {
  const int v = blockIdx.x * 256 + threadIdx.x;
  if (v >= nVar) return;
  float acc0 = b2[0], acc1 = b2[1], acc2 = b2[2], acc3 = b2[3];
  const float* h = H + (size_t)v * FMDIM;
  for (int k = 0; k < FMDIM; ++k) {
    const float hv = h[k];
    acc0 += hv * w2[k * 4 + 0];
    acc1 += hv * w2[k * 4 + 1];
    acc2 += hv * w2[k * 4 + 2];
    acc3 += hv * w2[k * 4 + 3];
  }
  float* so = outSig + (size_t)v * 4;
  so[0] = sigf(acc0); so[1] = sigf(acc1); so[2] = sigf(acc2); so[3] = sigf(acc3);
  if (outRaw) {
    float* ro = outRaw + (size_t)v * 4;
    ro[0] = acc0; ro[1] = acc1; ro[2] = acc2; ro[3] = acc3;
  }
}

// ---------------------------------------------------------------- launcher
extern "C" void kernel_launch(void* const* d_in, const int* in_sizes, int n_in,
                              void* d_out, int out_size, void* d_ws, size_t ws_size,
                              hipStream_t stream) {
  const int nCon = in_sizes[0];
  const int nVar = in_sizes[1];
  const int E    = in_sizes[3];

  const float* cv   = (const float*)d_in[0];
  const float* om   = (const float*)d_in[1];
  const float* ev   = (const float*)d_in[3];
  const int*   evar = (const int*)d_in[4];
  const int*   econ = (const int*)d_in[5];

  auto P = [&](int i) { return (const float*)d_in[i]; };
  // params sorted: cu(6..9), out(10..13), qc(14..17), qo(18..21), vu(22..25); each b1,b2,w1,w2
  const float *cu_b1 = P(6),  *cu_b2 = P(7),  *cu_w1 = P(8),  *cu_w2 = P(9);
  const float *ot_b1 = P(10), *ot_b2 = P(11), *ot_w1 = P(12), *ot_w2 = P(13);
  const float *qc_b1 = P(14), *qc_b2 = P(15), *qc_w1 = P(16), *qc_w2 = P(17);
  const float *qo_b1 = P(18), *qo_b2 = P(19), *qo_w1 = P(20), *qo_w2 = P(21);
  const float *vu_b1 = P(22), *vu_b2 = P(23), *vu_w1 = P(24), *vu_w2 = P(25);

  float* ws = (float*)d_ws;
  float* V    = ws; ws += (size_t)nVar * FMDIM;
  float* C    = ws; ws += (size_t)nCon * FMDIM;
  float* Q    = ws; ws += (size_t)nVar * FMDIM;
  float* H    = ws; ws += (size_t)nVar * FMDIM;
  float* left = ws; ws += (size_t)nCon * FMDIM;
  float* cmsg = ws; ws += (size_t)nCon * 384;
  float* vmsg = ws; ws += (size_t)nVar * 640;
  float* sc   = ws; ws += nCon;
  float* sv   = ws; ws += nVar;

  float* outF = (float*)d_out;

  const int gV  = nVar / 128;            // gemm blocks over variable rows
  const int gC  = nCon / 128;            // gemm blocks over constraint rows
  const int gE  = (E + 7) / 8;           // spmm blocks (8 edges / block)
  auto blocks = [](int n) { return (n + 255) / 256; };

  // --- one-time setup
  k_fill<<<blocks(nVar * FMDIM), 256, 0, stream>>>(V, nVar * FMDIM, 1.0f);
  k_fill<<<blocks(nCon * FMDIM), 256, 0, stream>>>(C, nCon * FMDIM, 1.0f);
  k_fill<<<blocks(nCon), 256, 0, stream>>>(sc, nCon, 0.0f);
  k_fill<<<blocks(nVar), 256, 0, stream>>>(sv, nVar, 0.0f);
  k_scalers<<<blocks(E), 256, 0, stream>>>(ev, econ, evar, sc, sv, E);

  for (int t = 0; t < NPASS; ++t) {
    // const_query = mlp_qc(V); sig_q -> Q
    k_gemm_wmma<<<gV, 256, 0, stream>>>(V, FMDIM, FMDIM, qc_w1, qc_b1, H, FMDIM,
                                        ACT_RELU, nullptr, nullptr, 0, 0.0f);
    k_gemm_wmma<<<gV, 256, 0, stream>>>(H, FMDIM, FMDIM, qc_w2, qc_b2, Q, FMDIM,
                                        ACT_SIG, nullptr, nullptr, 0, 0.0f);
    // left = spmm(var -> con, sig_q)
    k_fill<<<blocks(nCon * FMDIM), 256, 0, stream>>>(left, nCon * FMDIM, 0.0f);
    k_spmm<<<gE, 256, 0, stream>>>(ev, evar, econ, Q, FMDIM, left, FMDIM, E);
    // var_msg scaffolding (old V, zeros for c2v/const_grad, om broadcast)
    k_vmsg_init<<<blocks(nVar * FMDIM), 256, 0, stream>>>(vmsg, V, om, nVar);
    // const_msg = [C | relu(left-cv)/sc | relu(cv-left)/sc]
    k_cmsg<<<blocks(nCon * FMDIM), 256, 0, stream>>>(cmsg, C, left, cv, sc, nCon);
    // constraints = mlp_cu(const_msg, norm) + 0.5*C
    k_gemm_wmma<<<gC, 256, 0, stream>>>(cmsg, 384, 384, cu_w1, cu_b1, H, FMDIM,
                                        ACT_NONE, nullptr, nullptr, 0, 0.0f);
    k_norm_relu<<<(nCon + 7) / 8, 256, 0, stream>>>(H, nCon);
    k_gemm_wmma<<<gC, 256, 0, stream>>>(H, FMDIM, FMDIM, cu_w2, cu_b2, C, FMDIM,
                                        ACT_NONE, nullptr, C, FMDIM, 0.5f);
    // const_grad (act from left/cv) -> var_msg[:,384:512]
    k_spmm_act<<<gE, 256, 0, stream>>>(ev, econ, evar, left, cv, vmsg, E);
    // obj_loss = sigmoid(mlp_qo(V)) * om -> var_msg[:,256:384]
    k_gemm_wmma<<<gV, 256, 0, stream>>>(V, FMDIM, FMDIM, qo_w1, qo_b1, H, FMDIM,
                                        ACT_RELU, nullptr, nullptr, 0, 0.0f);
    k_gemm_wmma<<<gV, 256, 0, stream>>>(H, FMDIM, FMDIM, qo_w2, qo_b2, vmsg + 256, 640,
                                        ACT_SIG, om, nullptr, 0, 0.0f);
    // c2v = spmm(con -> var, new C) / sv -> var_msg[:,128:256]
    k_spmm<<<gE, 256, 0, stream>>>(ev, econ, evar, C, FMDIM, vmsg + 128, 640, E);
    k_c2v_div<<<blocks(nVar * FMDIM), 256, 0, stream>>>(vmsg, sv, nVar);
    // variables = mlp_vu(var_msg, norm) + 0.5*V
    k_gemm_wmma<<<gV, 256, 0, stream>>>(vmsg, 640, 640, vu_w1, vu_b1, H, FMDIM,
                                        ACT_NONE, nullptr, nullptr, 0, 0.0f);
    k_norm_relu<<<(nVar + 7) / 8, 256, 0, stream>>>(H, nVar);
    k_gemm_wmma<<<gV, 256, 0, stream>>>(H, FMDIM, FMDIM, vu_w2, vu_b2, V, FMDIM,
                                        ACT_NONE, nullptr, V, FMDIM, 0.5f);
    // out_vars = mlp_out(V, norm); outputs[t] = sigmoid(out_vars)
    k_gemm_wmma<<<gV, 256, 0, stream>>>(V, FMDIM, FMDIM, ot_w1, ot_b1, H, FMDIM,
                                        ACT_NONE, nullptr, nullptr, 0, 0.0f);
    k_norm_relu<<<(nVar + 7) / 8, 256, 0, stream>>>(H, nVar);
    k_out_head<<<blocks(nVar), 256, 0, stream>>>(
        H, ot_w2, ot_b2, outF + (size_t)t * nVar * OUTB,
        (t == NPASS - 1) ? (outF + (size_t)NPASS * nVar * OUTB) : nullptr, nVar);
  }
}